// MoE_68839735821022
// MI455X (gfx1250) — compile-verified
//
#include <hip/hip_runtime.h>
#include <hip/hip_bf16.h>

// ---------------------------------------------------------------------------
// MoE (router + quantile sparsify + topk + diag path + per-expert LoRA) for
// MI455X / gfx1250.  All GEMMs use V_WMMA_F32_16X16X4_F32 (fp32, matching the
// fp32 reference).  Global->LDS staging uses CDNA5 async-to-LDS copies
// (ASYNCcnt) with LDS double buffering so DMA overlaps the WMMA bursts.
// ---------------------------------------------------------------------------

constexpr int DIM  = 1024;
constexpr int HID  = 4096;
constexpr int E    = 128;
constexpr int TOPK = 4;
constexpr int R    = 128;
constexpr int CAP  = 256;
constexpr int N    = 4096;          // BS * SEQ
constexpr int M    = N * TOPK;      // 16384 assignments
constexpr int NTOT = N * E;         // 524288 logits
// jnp.quantile(|logits|, 0.8, 'linear'): pos = 0.8*(NTOT-1) = 419429.6
constexpr int   RANK0 = 419429;
constexpr float QFRAC = 0.6f;

typedef __attribute__((ext_vector_type(2))) float v2f;
typedef __attribute__((ext_vector_type(8))) float v8f;
typedef int i32x4 __attribute__((vector_size(16)));   // matches builtin proto

// ----- shared WMMA-f32 GEMM core: 64x64 tile, 4 waves, BK=32, 2x LDS buf ---
#define BM 64
#define BN 64
#define BK 32
#define LPAD 36             // 16B-aligned rows; stride 36 => conflict-free
#define TILE_F (BM * LPAD)  // floats per LDS tile

#if __has_builtin(__builtin_amdgcn_global_load_async_to_lds_b128)
#define ASYNC_LDS 1
#endif

__device__ __forceinline__ void lds_async_wait() {
#if defined(ASYNC_LDS)
#if __has_builtin(__builtin_amdgcn_s_wait_asynccnt)
  __builtin_amdgcn_s_wait_asynccnt(0);
#else
  asm volatile("s_wait_asynccnt 0x0" ::: "memory");
#endif
#endif
}

// copy 16 bytes global -> LDS (async DMA when available)
__device__ __forceinline__ void copy16_to_lds(const float* g, float* l) {
#if defined(ASYNC_LDS)
  __builtin_amdgcn_global_load_async_to_lds_b128(
      (__attribute__((address_space(1))) i32x4*)g,
      (__attribute__((address_space(3))) i32x4*)l,
      0, 0);
#else
  const float4 v = *(const float4*)g;
  l[0] = v.x; l[1] = v.y; l[2] = v.z; l[3] = v.w;
#endif
}

__device__ __forceinline__ void gemm_core_f32(
    const float* const aRow[4], const float* const bRow[4],
    int kTotal, float* As, float* Bs, v8f acc[2][2],
    int lane, int wm, int wn)
{
  const int tid   = threadIdx.x;
  const int lhalf = lane >> 4;      // 0: lanes 0-15, 1: lanes 16-31
  const int lmod  = lane & 15;

  // issue one 64x32 A tile + 64x32 B tile into LDS buffer `buf`
  auto loadTile = [&](int k0, int buf) {
#pragma unroll
    for (int i = 0; i < 4; ++i) {
      const int linear = i * 128 + tid;
      const int r  = linear >> 3;          // 8 float4 per 32-float row
      const int cc = (linear & 7) * 4;
      copy16_to_lds(aRow[i] + k0 + cc, &As[buf * TILE_F + r * LPAD + cc]);
      copy16_to_lds(bRow[i] + k0 + cc, &Bs[buf * TILE_F + r * LPAD + cc]);
    }
  };

  loadTile(0, 0);                          // prologue: tile 0 in flight
  int buf = 0;
  for (int k0 = 0; k0 < kTotal; k0 += BK, buf ^= 1) {
    lds_async_wait();                      // my async writes (incl. cur) done
    __syncthreads();                       // everyone's tile visible
    if (k0 + BK < kTotal) loadTile(k0 + BK, buf ^ 1);   // overlap next DMA
    const float* Ab = &As[buf * TILE_F];
    const float* Bb = &Bs[buf * TILE_F];
#pragma unroll
    for (int kk = 0; kk < BK; kk += 4) {
      v2f a[2], b[2];
#pragma unroll
      for (int f = 0; f < 2; ++f) {
        const int ar = wm * 32 + f * 16 + lmod;   // M rows of A fragment
        a[f][0] = Ab[ar * LPAD + kk + 2 * lhalf + 0];   // ISA: VGPR0=K0|K2
        a[f][1] = Ab[ar * LPAD + kk + 2 * lhalf + 1];   //      VGPR1=K1|K3
        const int br_ = wn * 32 + f * 16 + lmod;  // N cols (Bs is [n][k])
        b[f][0] = Bb[br_ * LPAD + kk + 2 * lhalf + 0];
        b[f][1] = Bb[br_ * LPAD + kk + 2 * lhalf + 1];
      }
#pragma unroll
      for (int fm = 0; fm < 2; ++fm)
#pragma unroll
        for (int fn = 0; fn < 2; ++fn)
          acc[fm][fn] = __builtin_amdgcn_wmma_f32_16x16x4_f32(
              false, a[fm], false, b[fn], (short)0, acc[fm][fn], false, false);
    }
  }
}

// ----- 1) router GEMM: logits = x @ Wr^T + br ------------------------------
__global__ __launch_bounds__(128) void k_router(
    const float* __restrict__ x, const float* __restrict__ Wr,
    const float* __restrict__ br, float* __restrict__ logits)
{
  __shared__ float As[2 * TILE_F];
  __shared__ float Bs[2 * TILE_F];
  const int mBase = blockIdx.y * BM, nBase = blockIdx.x * BN;
  const int tid = threadIdx.x, lane = tid & 31, wid = tid >> 5;
  const int wm = wid >> 1, wn = wid & 1;
  const float* aRow[4]; const float* bRow[4];
#pragma unroll
  for (int i = 0; i < 4; ++i) {
    const int row = (i * 128 + tid) >> 3;
    aRow[i] = x  + (size_t)(mBase + row) * DIM;
    bRow[i] = Wr + (size_t)(nBase + row) * DIM;
  }
  v8f z = {0.f,0.f,0.f,0.f,0.f,0.f,0.f,0.f};
  v8f acc[2][2] = {{z, z}, {z, z}};
  gemm_core_f32(aRow, bRow, DIM, As, Bs, acc, lane, wm, wn);
  const int lhalf = lane >> 4, lmod = lane & 15;
#pragma unroll
  for (int fm = 0; fm < 2; ++fm)
#pragma unroll
    for (int fn = 0; fn < 2; ++fn)
#pragma unroll
      for (int v = 0; v < 8; ++v) {
        const int row = mBase + wm * 32 + fm * 16 + v + lhalf * 8;
        const int col = nBase + wn * 32 + fn * 16 + lmod;
        logits[(size_t)row * E + col] = acc[fm][fn][v] + br[col];
      }
}

// ----- 2) exact 0.8-quantile of |logits| via 3-pass radix select -----------
__global__ __launch_bounds__(1024) void k_select_thr(
    const float* __restrict__ logits, float* __restrict__ thr)
{
  __shared__ int hist[2048];
  __shared__ int hist2[1024];
  __shared__ int s_hi, s_mid, s_rem0, s_rem1;
  const int tid = threadIdx.x;
  for (int i = tid; i < 2048; i += 1024) hist[i] = 0;
  __syncthreads();
  for (int i = tid; i < NTOT; i += 1024) {
    const unsigned u = __float_as_uint(logits[i]) & 0x7fffffffu;
    atomicAdd(&hist[u >> 20], 1);
  }
  __syncthreads();
  float v[2] = {0.f, 0.f};
  for (int t = 0; t < 2; ++t) {
    const int rank = RANK0 + t;
    if (tid == 0) {
      int cum = 0, b = 0;
      for (; b < 2048; ++b) { if (cum + hist[b] > rank) break; cum += hist[b]; }
      s_hi = b; s_rem0 = rank - cum;
    }
    __syncthreads();
    const int hi = s_hi, rem0 = s_rem0;
    if (tid < 1024) hist2[tid] = 0;
    __syncthreads();
    for (int i = tid; i < NTOT; i += 1024) {
      const unsigned u = __float_as_uint(logits[i]) & 0x7fffffffu;
      if ((int)(u >> 20) == hi) atomicAdd(&hist2[(u >> 10) & 1023], 1);
    }
    __syncthreads();
    if (tid == 0) {
      int cum = 0, b = 0;
      for (; b < 1024; ++b) { if (cum + hist2[b] > rem0) break; cum += hist2[b]; }
      s_mid = b; s_rem1 = rem0 - cum;
    }
    __syncthreads();
    const int mid = s_mid, rem1 = s_rem1;
    const unsigned pref = ((unsigned)hi << 10) | (unsigned)mid;
    if (tid < 1024) hist2[tid] = 0;
    __syncthreads();
    for (int i = tid; i < NTOT; i += 1024) {
      const unsigned u = __float_as_uint(logits[i]) & 0x7fffffffu;
      if ((u >> 10) == pref) atomicAdd(&hist2[u & 1023], 1);
    }
    __syncthreads();
    if (tid == 0) {
      int cum = 0, b = 0;
      for (; b < 1024; ++b) { if (cum + hist2[b] > rem1) break; cum += hist2[b]; }
      v[t] = __uint_as_float((pref << 10) | (unsigned)b);
    }
    __syncthreads();
  }
  if (tid == 0) thr[0] = (1.0f - QFRAC) * v[0] + QFRAC * v[1];
}

// ----- 3) per-token top-4 (+threshold) + softmax; one wave32 per token -----
__global__ __launch_bounds__(256) void k_topk(
    const float* __restrict__ logits, const float* __restrict__ thrP,
    int* __restrict__ topi, float* __restrict__ scores)
{
  const int lane = threadIdx.x & 31, wid = threadIdx.x >> 5;
  const int n = blockIdx.x * 8 + wid;
  const float thr = thrP[0];
  float vals[4];
#pragma unroll
  for (int j = 0; j < 4; ++j) {
    const float vv = logits[(size_t)n * E + lane + 32 * j];
    vals[j] = (fabsf(vv) < thr) ? 0.0f : vv;   // magnitude-quantile sparsify
  }
  float tv[4]; int ti[4];
#pragma unroll
  for (int k = 0; k < 4; ++k) {
    float bestv = -__builtin_inff(); int besti = 0x7fffffff;
#pragma unroll
    for (int j = 0; j < 4; ++j) {
      const int idx = lane + 32 * j;
      if (vals[j] > bestv || (vals[j] == bestv && idx < besti)) { bestv = vals[j]; besti = idx; }
    }
    for (int off = 16; off > 0; off >>= 1) {       // wave32 argmax reduce
      const float ov = __shfl_xor(bestv, off, 32);
      const int   oi = __shfl_xor(besti, off, 32);
      if (ov > bestv || (ov == bestv && oi < besti)) { bestv = ov; besti = oi; }
    }
    tv[k] = bestv; ti[k] = besti;
    const int jj = besti >> 5, owner = besti & 31;
#pragma unroll
    for (int j = 0; j < 4; ++j)
      if (j == jj && lane == owner) vals[j] = -__builtin_inff();
  }
  const float mx = tv[0];
  const float e0 = __expf(tv[0] - mx), e1 = __expf(tv[1] - mx),
              e2 = __expf(tv[2] - mx), e3 = __expf(tv[3] - mx);
  const float inv = 1.0f / (e0 + e1 + e2 + e3);
  if (lane == 0) {
    topi[n * 4 + 0] = ti[0]; topi[n * 4 + 1] = ti[1];
    topi[n * 4 + 2] = ti[2]; topi[n * 4 + 3] = ti[3];
    scores[n * 4 + 0] = e0 * inv; scores[n * 4 + 1] = e1 * inv;
    scores[n * 4 + 2] = e2 * inv; scores[n * 4 + 3] = e3 * inv;
  }
}

// ----- 4) xeff = x * sum_k s_k * diag[e_k] ---------------------------------
__global__ __launch_bounds__(256) void k_xeff(
    const float* __restrict__ x, const float* __restrict__ dg,
    const int* __restrict__ topi, const float* __restrict__ scores,
    float* __restrict__ xeff)
{
  const int n = blockIdx.x, t = threadIdx.x;         // 256 threads * float4
  int   ei[4]; float s[4];
#pragma unroll
  for (int k = 0; k < 4; ++k) { ei[k] = topi[n * 4 + k]; s[k] = scores[n * 4 + k]; }
  const float4 xv = ((const float4*)(x + (size_t)n * DIM))[t];
  float4 eff = {0.f, 0.f, 0.f, 0.f};
#pragma unroll
  for (int k = 0; k < 4; ++k) {
    const float4 dv = ((const float4*)(dg + (size_t)ei[k] * DIM))[t];
    eff.x += s[k] * dv.x; eff.y += s[k] * dv.y;
    eff.z += s[k] * dv.z; eff.w += s[k] * dv.w;
  }
  float4 o = {xv.x * eff.x, xv.y * eff.y, xv.z * eff.z, xv.w * eff.w};
  ((float4*)(xeff + (size_t)n * DIM))[t] = o;
}

// ----- 5) deterministic capacity dispatch: one wave per expert -------------
__global__ __launch_bounds__(128) void k_build_rows(
    const int* __restrict__ topi, const float* __restrict__ scores,
    int* __restrict__ r2t, float* __restrict__ rscale, int* __restrict__ cnt)
{
  const int gw = (int)(blockIdx.x * blockDim.x + threadIdx.x) >> 5;
  const int lane = threadIdx.x & 31;
  if (gw >= E) return;
  const int ex = gw;
  for (int i = lane; i < CAP; i += 32) { r2t[ex * CAP + i] = 0; rscale[ex * CAP + i] = 0.f; }
  int count = 0;
  for (int m0 = 0; m0 < M; m0 += 32) {              // assignment-order scan ==
    const int m = m0 + lane;                        // reference cumsum positions
    const bool match = (topi[m] == ex);
    const unsigned msk = (unsigned)__ballot(match);
    const int pos = count + __popc(msk & ((1u << lane) - 1u));
    if (match && pos < CAP) {
      r2t[ex * CAP + pos] = m >> 2;                 // source token
      rscale[ex * CAP + pos] = scores[m];
    }
    count += __popc(msk);
  }
  if (lane == 0) cnt[ex] = count;
}

// ----- 6) diag GEMM: out = xeff @ Wp^T + bp --------------------------------
__global__ __launch_bounds__(128) void k_diag(
    const float* __restrict__ xeff, const float* __restrict__ Wp,
    const float* __restrict__ bp, float* __restrict__ out)
{
  __shared__ float As[2 * TILE_F];
  __shared__ float Bs[2 * TILE_F];
  const int mBase = blockIdx.y * BM, nBase = blockIdx.x * BN;
  const int tid = threadIdx.x, lane = tid & 31, wid = tid >> 5;
  const int wm = wid >> 1, wn = wid & 1;
  const float* aRow[4]; const float* bRow[4];
#pragma unroll
  for (int i = 0; i < 4; ++i) {
    const int row = (i * 128 + tid) >> 3;
    aRow[i] = xeff + (size_t)(mBase + row) * DIM;
    bRow[i] = Wp   + (size_t)(nBase + row) * DIM;
  }
  v8f z = {0.f,0.f,0.f,0.f,0.f,0.f,0.f,0.f};
  v8f acc[2][2] = {{z, z}, {z, z}};
  gemm_core_f32(aRow, bRow, DIM, As, Bs, acc, lane, wm, wn);
  const int lhalf = lane >> 4, lmod = lane & 15;
#pragma unroll
  for (int fm = 0; fm < 2; ++fm)
#pragma unroll
    for (int fn = 0; fn < 2; ++fn)
#pragma unroll
      for (int v = 0; v < 8; ++v) {
        const int row = mBase + wm * 32 + fm * 16 + v + lhalf * 8;
        const int col = nBase + wn * 32 + fn * 16 + lmod;
        out[(size_t)row * HID + col] = acc[fm][fn][v] + bp[col];  // scores sum==1
      }
}

// ----- 7) per-expert LoRA down: h = relu(gather(x) @ W1[e]^T + b1[e]) ------
__global__ __launch_bounds__(128) void k_hgemm(
    const float* __restrict__ xf, const float* __restrict__ W1,
    const float* __restrict__ b1, const int* __restrict__ r2t,
    const int* __restrict__ cnt, float* __restrict__ hbuf)
{
  const int e = blockIdx.z;
  const int cc = cnt[e] < CAP ? cnt[e] : CAP;
  const int mBase = blockIdx.y * BM;
  if (mBase >= cc) return;                      // skip empty row tiles
  __shared__ float As[2 * TILE_F];
  __shared__ float Bs[2 * TILE_F];
  const int nBase = blockIdx.x * BN;
  const int tid = threadIdx.x, lane = tid & 31, wid = tid >> 5;
  const int wm = wid >> 1, wn = wid & 1;
  const float* aRow[4]; const float* bRow[4];
#pragma unroll
  for (int i = 0; i < 4; ++i) {
    const int row = (i * 128 + tid) >> 3;
    const int tok = r2t[e * CAP + mBase + row];          // gathered dispatch
    aRow[i] = xf + (size_t)tok * DIM;
    bRow[i] = W1 + (size_t)e * R * DIM + (size_t)(nBase + row) * DIM;
  }
  v8f z = {0.f,0.f,0.f,0.f,0.f,0.f,0.f,0.f};
  v8f acc[2][2] = {{z, z}, {z, z}};
  gemm_core_f32(aRow, bRow, DIM, As, Bs, acc, lane, wm, wn);
  const int lhalf = lane >> 4, lmod = lane & 15;
#pragma unroll
  for (int fm = 0; fm < 2; ++fm)
#pragma unroll
    for (int fn = 0; fn < 2; ++fn)
#pragma unroll
      for (int v = 0; v < 8; ++v) {
        const int row = mBase + wm * 32 + fm * 16 + v + lhalf * 8;
        const int col = nBase + wn * 32 + fn * 16 + lmod;
        const float hv = acc[fm][fn][v] + b1[e * R + col];
        hbuf[((size_t)e * CAP + row) * R + col] = hv > 0.f ? hv : 0.f;
      }
}

// ----- 8) per-expert LoRA up fused with scaled scatter-add combine ---------
__global__ __launch_bounds__(128) void k_ygemm_scatter(
    const float* __restrict__ hbuf, const float* __restrict__ W2,
    const float* __restrict__ b2, const int* __restrict__ r2t,
    const float* __restrict__ rscale, const int* __restrict__ cnt,
    float* __restrict__ out)
{
  const int e = blockIdx.z;
  const int cc = cnt[e] < CAP ? cnt[e] : CAP;
  const int mBase = blockIdx.y * BM;
  if (mBase >= cc) return;
  __shared__ float As[2 * TILE_F];
  __shared__ float Bs[2 * TILE_F];
  const int nBase = blockIdx.x * BN;
  const int tid = threadIdx.x, lane = tid & 31, wid = tid >> 5;
  const int wm = wid >> 1, wn = wid & 1;
  const float* aRow[4]; const float* bRow[4];
#pragma unroll
  for (int i = 0; i < 4; ++i) {
    const int row = (i * 128 + tid) >> 3;
    aRow[i] = hbuf + ((size_t)e * CAP + mBase + row) * R;
    bRow[i] = W2   + (size_t)e * HID * R + (size_t)(nBase + row) * R;
  }
  v8f z = {0.f,0.f,0.f,0.f,0.f,0.f,0.f,0.f};
  v8f acc[2][2] = {{z, z}, {z, z}};
  gemm_core_f32(aRow, bRow, R, As, Bs, acc, lane, wm, wn);
  const int lhalf = lane >> 4, lmod = lane & 15;
#pragma unroll
  for (int fm = 0; fm < 2; ++fm)
#pragma unroll
    for (int fn = 0; fn < 2; ++fn)
#pragma unroll
      for (int v = 0; v < 8; ++v) {
        const int row = mBase + wm * 32 + fm * 16 + v + lhalf * 8;
        const int col = nBase + wn * 32 + fn * 16 + lmod;
        const int tok = r2t[e * CAP + row];
        const float s = rscale[e * CAP + row];     // 0 for padding/dropped
        const float val = (acc[fm][fn][v] + b2[(size_t)e * HID + col]) * s;
        atomicAdd(&out[(size_t)tok * HID + col], val);
      }
}

// ---------------------------------------------------------------------------
extern "C" void kernel_launch(void* const* d_in, const int* in_sizes, int n_in,
                              void* d_out, int out_size, void* d_ws, size_t ws_size,
                              hipStream_t stream) {
  (void)in_sizes; (void)n_in; (void)out_size; (void)ws_size;
  const float* x   = (const float*)d_in[0];
  const float* Wr  = (const float*)d_in[1];
  const float* br  = (const float*)d_in[2];
  const float* dg  = (const float*)d_in[3];
  const float* Wp  = (const float*)d_in[4];
  const float* bp  = (const float*)d_in[5];
  const float* W1  = (const float*)d_in[6];
  const float* b1  = (const float*)d_in[7];
  const float* W2  = (const float*)d_in[8];
  const float* b2  = (const float*)d_in[9];
  float* out = (float*)d_out;

  char* ws = (char*)d_ws;
  size_t off = 0;
  auto alloc = [&](size_t bytes) {
    void* p = ws + off;
    off = (off + bytes + 255) & ~(size_t)255;
    return p;
  };
  float* logits = (float*)alloc((size_t)N * E * sizeof(float));        //  2 MB
  float* thr    = (float*)alloc(256);
  int*   topi   = (int*)  alloc((size_t)M * sizeof(int));              // 64 KB
  float* scores = (float*)alloc((size_t)M * sizeof(float));            // 64 KB
  int*   cnt    = (int*)  alloc((size_t)E * sizeof(int));
  int*   r2t    = (int*)  alloc((size_t)E * CAP * sizeof(int));        //128 KB
  float* rscale = (float*)alloc((size_t)E * CAP * sizeof(float));      //128 KB
  float* xeff   = (float*)alloc((size_t)N * DIM * sizeof(float));      // 16 MB
  float* hbuf   = (float*)alloc((size_t)E * CAP * R * sizeof(float));  // 16 MB

  // 1) router logits
  k_router<<<dim3(E / BN, N / BM), 128, 0, stream>>>(x, Wr, br, logits);
  // 2) quantile threshold (exact radix select on L2-resident logits)
  k_select_thr<<<1, 1024, 0, stream>>>(logits, thr);
  // 3) top-4 + softmax
  k_topk<<<N / 8, 256, 0, stream>>>(logits, thr, topi, scores);
  // 4) xeff = x * (sum_k s_k diag[e_k])
  k_xeff<<<N, 256, 0, stream>>>(x, dg, topi, scores, xeff);
  // 5) deterministic dispatch tables (matches reference cumsum/drop order)
  k_build_rows<<<(E * 32 + 127) / 128, 128, 0, stream>>>(topi, scores, r2t, rscale, cnt);
  // 6) diag path writes out = xeff @ Wp^T + bp   (initializes every element)
  k_diag<<<dim3(HID / BN, N / BM), 128, 0, stream>>>(xeff, Wp, bp, out);
  // 7) expert LoRA down (gathered A), tile-skipped by occupancy
  k_hgemm<<<dim3(R / BN, CAP / BM, E), 128, 0, stream>>>(x, W1, b1, r2t, cnt, hbuf);
  // 8) expert LoRA up + scaled scatter-add combine into out
  k_ygemm_scatter<<<dim3(HID / BN, CAP / BM, E), 128, 0, stream>>>(
      hbuf, W2, b2, r2t, rscale, cnt, out);
}